// Model_45629732553058
// MI455X (gfx1250) — compile-verified
//
#include <hip/hip_runtime.h>

typedef float v2f __attribute__((ext_vector_type(2)));
typedef float v8f __attribute__((ext_vector_type(8)));

#define W1_PITCH 516   // floats; 516 % 64 == 4 banks -> conflict-free strided b64 reads

__global__ __launch_bounds__(256) void mlp_topk_kernel(
    const float* __restrict__ x,  const float* __restrict__ W1, const float* __restrict__ b1,
    const float* __restrict__ W2, const float* __restrict__ b2,
    const float* __restrict__ W3, const float* __restrict__ b3,
    float* __restrict__ out)
{
    __shared__ float sW1[16 * W1_PITCH];  // W1 (8 x 512) zero-padded to 16 rows
    __shared__ float sH1[128 * 8];        // h1 for this block's 128 rows

    const int tid  = threadIdx.x;
    const int lane = tid & 31;
    const int wave = tid >> 5;                    // 0..7
    const int rowBlock = blockIdx.x * 128;

    // ---- stage W1 into padded LDS (rows 8..15 zeroed; coalesced float4) ----
    for (int i = tid; i < 16 * 128; i += 256) {   // 2048 float4 chunks
        int r  = i >> 7;
        int c4 = i & 127;
        float4 v = make_float4(0.f, 0.f, 0.f, 0.f);
        if (r < 8) v = ((const float4*)(W1 + r * 512))[c4];
        *(float4*)&sW1[r * W1_PITCH + c4 * 4] = v;
    }

    // ---- hoist uniform small weights (compiler -> s_load) ----
    float w2r[64];
    float b2r[8];
#pragma unroll
    for (int i = 0; i < 64; ++i) w2r[i] = W2[i];
#pragma unroll
    for (int i = 0; i < 8; ++i)  b2r[i] = b2[i];

    __syncthreads();

    // ---- layer 1 via V_WMMA_F32_16X16X4_F32, one 16-row tile per wave ----
    // A layout (16x4 f32): lane L(0..15)->M=L, K={k0,k0+1}; lane L+16->M=L, K={k0+2,k0+3}
    // B layout (4x16 f32): lane -> N, vgpr/half -> K (W1^T chunk; N>=8 rows are zero in LDS)
    const int mrow = lane & 15;
    const int hi   = lane >> 4;

    const float* xrow   = x + (size_t)(rowBlock + wave * 16 + mrow) * 512 + 2 * hi;
    const float* w1lane = &sW1[mrow * W1_PITCH + 2 * hi];

    v8f c = {};
#pragma unroll 8
    for (int kk = 0; kk < 128; ++kk) {
        v2f a = *(const v2f*)(xrow + kk * 4);       // global_load_b64
        v2f b = *(const v2f*)(w1lane + kk * 4);     // ds_load_b64 (pairs -> ds_load_2addr_b64)
        c = __builtin_amdgcn_wmma_f32_16x16x4_f32(
                /*neg_a=*/false, a, /*neg_b=*/false, b,
                /*c_mod=*/(short)0, c, /*reuse_a=*/false, /*reuse_b=*/false);
    }

    // epilogue: bias + relu, write h1 tile to LDS
    if (mrow < 8) {
        float bias = b1[mrow];
#pragma unroll
        for (int g = 0; g < 8; ++g) {
            float v = c[g] + bias;
            v = v > 0.0f ? v : 0.0f;
            sH1[(wave * 16 + g + 8 * hi) * 8 + mrow] = v;
        }
    }
    __syncthreads();

    // ---- per-row phase: one wave per row, 16 rows per wave ----
    for (int g = 0; g < 16; ++g) {
        const int rowInBlk = wave * 16 + g;
        const int row = rowBlock + rowInBlk;

        float h1v[8];
#pragma unroll
        for (int h = 0; h < 8; ++h) h1v[h] = sH1[rowInBlk * 8 + h];  // broadcast

        float h2v[8];
#pragma unroll
        for (int k = 0; k < 8; ++k) {
            float acc = b2r[k];
#pragma unroll
            for (int h = 0; h < 8; ++h) acc += w2r[k * 8 + h] * h1v[h];
            h2v[k] = acc > 0.0f ? acc : 0.0f;
        }

        // layer 3: lane covers cols lane + 32*j  (coalesced)
        float p[16];
#pragma unroll
        for (int j = 0; j < 16; ++j) {
            int cidx = lane + 32 * j;
            const float4* w3p = (const float4*)(W3 + (size_t)cidx * 8);
            float4 wa = w3p[0], wb = w3p[1];
            float acc = b3[cidx];
            acc += wa.x * h2v[0] + wa.y * h2v[1] + wa.z * h2v[2] + wa.w * h2v[3];
            acc += wb.x * h2v[4] + wb.y * h2v[5] + wb.z * h2v[6] + wb.w * h2v[7];
            p[j] = acc;
        }

        // softmax (faithful: exp(y - rowmax) / sum)
        float m = p[0];
#pragma unroll
        for (int j = 1; j < 16; ++j) m = fmaxf(m, p[j]);
#pragma unroll
        for (int off = 16; off > 0; off >>= 1) m = fmaxf(m, __shfl_xor(m, off, 32));

        float s = 0.0f;
#pragma unroll
        for (int j = 0; j < 16; ++j) { p[j] = __expf(p[j] - m); s += p[j]; }
#pragma unroll
        for (int off = 16; off > 0; off >>= 1) s += __shfl_xor(s, off, 32);
        float inv = 1.0f / s;
#pragma unroll
        for (int j = 0; j < 16; ++j) p[j] *= inv;

        if (lane == 0)  p[0]  = 1.0f;   // col 0
        if (lane == 31) p[15] = 1.0f;   // col 511

        // extract 16 maxima; thr = 16th largest. p >= 0 so -1 is a safe tombstone.
        float thr = 0.0f;
        for (int it = 0; it < 16; ++it) {
            float lm = p[0];
#pragma unroll
            for (int j = 1; j < 16; ++j) lm = fmaxf(lm, p[j]);
            float mm = lm;
#pragma unroll
            for (int off = 16; off > 0; off >>= 1) mm = fmaxf(mm, __shfl_xor(mm, off, 32));
            unsigned ball = __builtin_amdgcn_ballot_w32(lm == mm);
            int firstLane = __ffs(ball) - 1;
            if (lane == firstLane) {
                bool done = false;
#pragma unroll
                for (int j = 0; j < 16; ++j) {
                    bool take = (!done) && (p[j] == mm);
                    p[j] = take ? -1.0f : p[j];
                    done = done || take;
                }
            }
            thr = mm;
        }

        // mask: extracted entries (p<0) were top-16 -> 1; ties with thr also 1 (matches ref)
        float* orow = out + (size_t)row * 512;
#pragma unroll
        for (int j = 0; j < 16; ++j) {
            float v = (p[j] >= thr || p[j] < 0.0f) ? 1.0f : 0.0f;
            orow[lane + 32 * j] = v;
        }
    }
}

extern "C" void kernel_launch(void* const* d_in, const int* in_sizes, int n_in,
                              void* d_out, int out_size, void* d_ws, size_t ws_size,
                              hipStream_t stream) {
    (void)n_in; (void)out_size; (void)d_ws; (void)ws_size;
    const float* x  = (const float*)d_in[0];
    const float* W1 = (const float*)d_in[1];
    const float* b1 = (const float*)d_in[2];
    const float* W2 = (const float*)d_in[3];
    const float* b2 = (const float*)d_in[4];
    const float* W3 = (const float*)d_in[5];
    const float* b3 = (const float*)d_in[6];
    float* out = (float*)d_out;

    const int Brows = in_sizes[0] / 512;      // 65536
    dim3 grid(Brows / 128), block(256);
    mlp_topk_kernel<<<grid, block, 0, stream>>>(x, W1, b1, W2, b2, W3, b3, out);
}